// RefLocal8MH_34394098106943
// MI455X (gfx1250) — compile-verified
//
#include <hip/hip_runtime.h>
#include <hip/hip_bf16.h>
#include <math.h>

typedef __attribute__((ext_vector_type(2))) float v2f;
typedef __attribute__((ext_vector_type(8))) float v8f;

#define HNUM 4
#define CI   16
#define CDIM 64
#define NDIM 64   // heads * Ci
#define KNB  25
#define KR   2

// ---------------------------------------------------------------------------
// Projection GEMM via fp32 WMMA:
//   Y[M x 64] = X[M x 64] * W[64 x 64]
// W supplied as [heads][C][Ci] (column n = head*16 + d):
//   W[c][n] = Wt[(n>>4)*(64*16) + c*16 + (n&15)]
// One wave computes a 16(M) x 64(N) slab: the A fragment (activation stream,
// the only real memory traffic) is loaded ONCE per K-step and reused across 4
// N-tiles -> 4 independent WMMA accumulator chains, 64 v_wmma_f32_16x16x4_f32
// per wave. The 16 KB weight matrix stays cache-resident.
// A 16x4 frag (ISA 7.12.2): lanes 0-15 -> M=lane, {K+0,K+1}; lanes 16-31 ->
// M=lane-16, {K+2,K+3}. B 4x16 frag mirrors the K split with N=lane&15.
// C/D: vgpr i holds M = i + 8*(lane>=16), N = lane&15.
// ---------------------------------------------------------------------------
__global__ void proj_wmma(const float* __restrict__ X,
                          const float* __restrict__ Wt,
                          float* __restrict__ Y, int M)
{
    const int lane        = threadIdx.x & 31;
    const int waveInBlock = threadIdx.x >> 5;
    const int mtile       = blockIdx.x * (blockDim.x >> 5) + waveInBlock;
    if (mtile * 16 >= M) return;                     // wave-uniform: EXEC stays all-1

    const int row = lane & 15;                       // A row / B-C column within tile
    const int hi  = lane >> 4;                       // low/high half of wave
    const int kb  = hi * 2;                          // K sub-offset within group of 4
    const int m0  = mtile * 16;

    const float* arow = X + (size_t)(m0 + row) * CDIM;

    // Per-lane weight base for each of the 4 N-tiles (head = ntile here since
    // CI == 16): W[c][nt*16 + row] = Wt[nt*1024 + c*16 + row]
    const float* wb0 = Wt + 0 * (CDIM * CI) + row;
    const float* wb1 = Wt + 1 * (CDIM * CI) + row;
    const float* wb2 = Wt + 2 * (CDIM * CI) + row;
    const float* wb3 = Wt + 3 * (CDIM * CI) + row;

    v8f acc0 = {}, acc1 = {}, acc2 = {}, acc3 = {};
#pragma unroll
    for (int kk = 0; kk < CDIM; kk += 4) {
        const int k0 = kk + kb;
        v2f a;
        a.x = arow[k0];
        a.y = arow[k0 + 1];
        v2f b0, b1, b2, b3;
        b0.x = wb0[k0 * CI]; b0.y = wb0[(k0 + 1) * CI];
        b1.x = wb1[k0 * CI]; b1.y = wb1[(k0 + 1) * CI];
        b2.x = wb2[k0 * CI]; b2.y = wb2[(k0 + 1) * CI];
        b3.x = wb3[k0 * CI]; b3.y = wb3[(k0 + 1) * CI];
        acc0 = __builtin_amdgcn_wmma_f32_16x16x4_f32(false, a, false, b0,
                                                     (short)0, acc0, false, false);
        acc1 = __builtin_amdgcn_wmma_f32_16x16x4_f32(false, a, false, b1,
                                                     (short)0, acc1, false, false);
        acc2 = __builtin_amdgcn_wmma_f32_16x16x4_f32(false, a, false, b2,
                                                     (short)0, acc2, false, false);
        acc3 = __builtin_amdgcn_wmma_f32_16x16x4_f32(false, a, false, b3,
                                                     (short)0, acc3, false, false);
    }

    // store: vgpr i -> row m0 + i + hi*8, column nt*16 + (lane&15)
    float* yp = Y + (size_t)(m0 + hi * 8) * NDIM + row;
#pragma unroll
    for (int i = 0; i < 8; ++i) {
        float* r = yp + (size_t)i * NDIM;
        r[0]  = acc0[i];
        r[16] = acc1[i];
        r[32] = acc2[i];
        r[48] = acc3[i];
    }
}

// ---------------------------------------------------------------------------
// Attention: per position, per head: 25-neighbor 16-dim dots, softmax over the
// 25 logits (zero-padded borders give logit 0, still enter the softmax),
// then average the 4 heads. One thread per (b,h,w) position.
// ---------------------------------------------------------------------------
__device__ __forceinline__ float dot16(const float4 q0, const float4 q1,
                                       const float4 q2, const float4 q3,
                                       const float4* __restrict__ r4)
{
    float4 r0 = r4[0], r1 = r4[1], r2 = r4[2], r3 = r4[3];
    float v = q0.x * r0.x;
    v = fmaf(q0.y, r0.y, v); v = fmaf(q0.z, r0.z, v); v = fmaf(q0.w, r0.w, v);
    v = fmaf(q1.x, r1.x, v); v = fmaf(q1.y, r1.y, v);
    v = fmaf(q1.z, r1.z, v); v = fmaf(q1.w, r1.w, v);
    v = fmaf(q2.x, r2.x, v); v = fmaf(q2.y, r2.y, v);
    v = fmaf(q2.z, r2.z, v); v = fmaf(q2.w, r2.w, v);
    v = fmaf(q3.x, r3.x, v); v = fmaf(q3.y, r3.y, v);
    v = fmaf(q3.z, r3.z, v); v = fmaf(q3.w, r3.w, v);
    return v;
}

__global__ void attn_softmax(const float* __restrict__ Q,
                             const float* __restrict__ R,
                             float* __restrict__ Out,
                             int B_, int H_, int W_)
{
    const int pos   = blockIdx.x * blockDim.x + threadIdx.x;
    const int total = B_ * H_ * W_;
    if (pos >= total) return;

    const int w = pos % W_;
    const int t = pos / W_;
    const int h = t % H_;
    const int b = t / H_;

    float acc[KNB];
#pragma unroll
    for (int k = 0; k < KNB; ++k) acc[k] = 0.f;

    const float4* q4 = (const float4*)(Q + (size_t)pos * NDIM);

    for (int head = 0; head < HNUM; ++head) {
        const float4 q0 = q4[head * 4 + 0];
        const float4 q1 = q4[head * 4 + 1];
        const float4 q2 = q4[head * 4 + 2];
        const float4 q3 = q4[head * 4 + 3];

        float attn[KNB];
        float mx = -INFINITY;
#pragma unroll
        for (int dy = -KR; dy <= KR; ++dy) {
#pragma unroll
            for (int dx = -KR; dx <= KR; ++dx) {
                const int k  = (dy + KR) * 5 + (dx + KR);
                const int yy = h + dy;
                const int xx = w + dx;
                float v = 0.f;
                if (yy >= 0 && yy < H_ && xx >= 0 && xx < W_) {
                    const size_t np = ((size_t)b * H_ + yy) * W_ + xx;
                    const float4* r4 =
                        (const float4*)(R + np * NDIM) + head * 4;
                    v = dot16(q0, q1, q2, q3, r4);
                }
                attn[k] = v;
                mx = fmaxf(mx, v);
            }
        }

        float s = 0.f;
#pragma unroll
        for (int k = 0; k < KNB; ++k) {
            const float e = __expf(attn[k] - mx);
            attn[k] = e;
            s += e;
        }
        const float inv = 0.25f / s;   // mean over 4 heads folded in
#pragma unroll
        for (int k = 0; k < KNB; ++k) acc[k] += attn[k] * inv;
    }

    float* op = Out + (size_t)pos * KNB;
#pragma unroll
    for (int k = 0; k < KNB; ++k) op[k] = acc[k];
}

// ---------------------------------------------------------------------------
extern "C" void kernel_launch(void* const* d_in, const int* in_sizes, int n_in,
                              void* d_out, int out_size, void* d_ws, size_t ws_size,
                              hipStream_t stream)
{
    (void)in_sizes; (void)n_in; (void)out_size; (void)ws_size;

    const float* main_ = (const float*)d_in[0];   // [4,128,128,64]
    const float* ref_  = (const float*)d_in[1];   // [4,128,128,64]
    const float* wmain = (const float*)d_in[2];   // [4,64,16]
    const float* wref  = (const float*)d_in[3];   // [4,64,16]
    float* out = (float*)d_out;                   // [4,128,128,25]

    const int B_ = 4, H_ = 128, W_ = 128;
    const int M  = B_ * H_ * W_;                  // 65536 rows

    float* q = (float*)d_ws;                      // [M,64] fp32
    float* r = q + (size_t)M * NDIM;              // [M,64] fp32 (total 33.6 MB)

    // one wave per 16 rows (full N=64 per wave), 8 waves (256 thr) per block
    const int wavesTotal = M / 16;                // 4096
    const int blocks     = (wavesTotal + 7) / 8;  // 512
    proj_wmma<<<blocks, 256, 0, stream>>>(main_, wmain, q, M);
    proj_wmma<<<blocks, 256, 0, stream>>>(ref_,  wref,  r, M);

    attn_softmax<<<(M + 255) / 256, 256, 0, stream>>>(q, r, out, B_, H_, W_);
}